// TripletLossBatchAll_83769042141680
// MI455X (gfx1250) — compile-verified
//
#include <hip/hip_runtime.h>
#include <hip/hip_bf16.h>

typedef __attribute__((ext_vector_type(2))) float v2f;
typedef __attribute__((ext_vector_type(8))) float v8f;

#define BATCH 512
#define DIM   128
#define TMARGIN 0.3f

// ---------------------------------------------------------------------------
// Kernel 0: row squared norms  sq[i] = sum_k e[i,k]^2
// ---------------------------------------------------------------------------
__global__ void sqnorm_kernel(const float* __restrict__ e, float* __restrict__ sq) {
    int i = blockIdx.x * blockDim.x + threadIdx.x;
    if (i >= BATCH) return;
    const float* row = e + (size_t)i * DIM;
    float s = 0.0f;
#pragma unroll 8
    for (int k = 0; k < DIM; ++k) s = fmaf(row[k], row[k], s);
    sq[i] = s;
}

// ---------------------------------------------------------------------------
// Kernel 1: dist[i,j] = safe_sqrt(sq[i] + sq[j] - 2*(e@e.T)[i,j])
// One wave (32 lanes) per 16x16 output tile, K accumulated with
// V_WMMA_F32_16X16X4_F32 (full-precision f32 matrix path on gfx1250).
//
// f32 A 16x4 layout: lane l, m = l&15, half = l>>4;
//   VGPR0 = A[m, k+2*half], VGPR1 = A[m, k+2*half+1]   (contiguous pair)
// f32 B 4x16 layout mirrors with n = l&15.
// C/D: VGPR v holds C[v + 8*half, l&15].
// ---------------------------------------------------------------------------
__global__ void dist_wmma_kernel(const float* __restrict__ e,
                                 const float* __restrict__ sq,
                                 float* __restrict__ dist) {
    const int i0   = blockIdx.x * 16;
    const int j0   = blockIdx.y * 16;
    const int lane = threadIdx.x;      // 0..31, full wave, no divergence
    const int half = lane >> 4;        // 0 or 1
    const int mn   = lane & 15;

    const float* __restrict__ arow = e + (size_t)(i0 + mn) * DIM + 2 * half;
    const float* __restrict__ brow = e + (size_t)(j0 + mn) * DIM + 2 * half;

    v8f acc = {};
#pragma unroll 4
    for (int k = 0; k < DIM; k += 4) {
        v2f a = *(const v2f*)(arow + k);   // 8B-aligned float2
        v2f b = *(const v2f*)(brow + k);
        acc = __builtin_amdgcn_wmma_f32_16x16x4_f32(
            /*neg_a=*/false, a, /*neg_b=*/false, b,
            /*c_mod=*/(short)0, acc, /*reuse_a=*/false, /*reuse_b=*/false);
    }

    const int   n   = mn;
    const float sqn = sq[j0 + n];
#pragma unroll
    for (int v = 0; v < 8; ++v) {
        const int m  = v + 8 * half;
        float d2 = sq[i0 + m] + sqn - 2.0f * acc[v];
        d2 = fmaxf(d2, 0.0f);
        dist[(size_t)(i0 + m) * BATCH + (j0 + n)] = (d2 > 0.0f) ? sqrtf(d2) : 0.0f;
    }
}

// ---------------------------------------------------------------------------
// Kernel 2: per-anchor masked triplet scan.
// One block per anchor; anchor's dist row + labels cached in LDS.
// sum  += relu(d[a,p] - d[a,n] + margin)   over valid (p,n)
// cnt  += (value > 1e-16)                  (exact integer count)
// ---------------------------------------------------------------------------
__global__ void triplet_kernel(const float* __restrict__ dist,
                               const int*   __restrict__ labels,
                               float*        __restrict__ psum,
                               unsigned int* __restrict__ pcnt) {
    const int a   = blockIdx.x;
    const int tid = threadIdx.x;        // 0..255

    __shared__ float        drow[BATCH];
    __shared__ int          lab[BATCH];
    __shared__ float        ssum[256];
    __shared__ unsigned int scnt[256];

    for (int i = tid; i < BATCH; i += 256) {
        drow[i] = dist[(size_t)a * BATCH + i];
        lab[i]  = labels[i];
    }
    __syncthreads();

    const int la = lab[a];
    float        sum = 0.0f;
    unsigned int cnt = 0u;

    for (int p = 0; p < BATCH; ++p) {
        if (p == a || lab[p] != la) continue;      // anchor-positive mask
        const float dpm = drow[p] + TMARGIN;
        for (int n = tid; n < BATCH; n += 256) {
            if (lab[n] == la) continue;            // anchor-negative mask
            const float v = dpm - drow[n];
            sum += fmaxf(v, 0.0f);
            cnt += (v > 1e-16f) ? 1u : 0u;
        }
    }

    ssum[tid] = sum;
    scnt[tid] = cnt;
    __syncthreads();
    for (int s = 128; s > 0; s >>= 1) {
        if (tid < s) { ssum[tid] += ssum[tid + s]; scnt[tid] += scnt[tid + s]; }
        __syncthreads();
    }
    if (tid == 0) { psum[a] = ssum[0]; pcnt[a] = scnt[0]; }
}

// ---------------------------------------------------------------------------
// Kernel 3: final deterministic reduction + divide.
// ---------------------------------------------------------------------------
__global__ void finalize_kernel(const float*        __restrict__ psum,
                                const unsigned int* __restrict__ pcnt,
                                float*              __restrict__ out) {
    __shared__ float              ssum[BATCH];
    __shared__ unsigned long long scnt[BATCH];
    const int tid = threadIdx.x;
    ssum[tid] = psum[tid];
    scnt[tid] = (unsigned long long)pcnt[tid];
    __syncthreads();
    for (int s = BATCH / 2; s > 0; s >>= 1) {
        if (tid < s) { ssum[tid] += ssum[tid + s]; scnt[tid] += scnt[tid + s]; }
        __syncthreads();
    }
    if (tid == 0) {
        const float  total = ssum[0];
        const double np    = (double)scnt[0];
        const float  denom = (np > 0.0) ? (float)(np + 1e-16) : 1.0f;
        out[0] = total / denom;
    }
}

// ---------------------------------------------------------------------------
// Launch
// ---------------------------------------------------------------------------
extern "C" void kernel_launch(void* const* d_in, const int* in_sizes, int n_in,
                              void* d_out, int out_size, void* d_ws, size_t ws_size,
                              hipStream_t stream) {
    const float* e      = (const float*)d_in[0];   // (512,128) f32
    const int*   labels = (const int*)d_in[1];     // (512,)    i32
    float*       out    = (float*)d_out;           // scalar f32

    char*  ws   = (char*)d_ws;
    float* dist = (float*)ws;                                       // 512*512 f32 = 1 MB
    float* sq   = (float*)(ws + (size_t)BATCH * BATCH * sizeof(float));
    float* psum = sq + BATCH;
    unsigned int* pcnt = (unsigned int*)(psum + BATCH);

    sqnorm_kernel   <<<(BATCH + 255) / 256, 256, 0, stream>>>(e, sq);
    dist_wmma_kernel<<<dim3(BATCH / 16, BATCH / 16), 32, 0, stream>>>(e, sq, dist);
    triplet_kernel  <<<BATCH, 256, 0, stream>>>(dist, labels, psum, pcnt);
    finalize_kernel <<<1, BATCH, 0, stream>>>(psum, pcnt, out);
}